// LocallyConnected2d_18631568130760
// MI455X (gfx1250) — compile-verified
//
#include <hip/hip_runtime.h>
#include <stdint.h>

typedef __attribute__((ext_vector_type(2))) float v2f;
typedef __attribute__((ext_vector_type(8))) float v8f;

#define B_        8
#define CIN_      64
#define COUT_     64
#define KTAPS     9
#define KDIM      576                 // CIN * 9, c-major: K = c*9 + tap
#define WSTRIDE_C 9216                // 1024*9 floats between c rows of weight
#define WSTRIDE_O 589824              // 64*1024*9 floats between cout rows
#define GROUPK    36                  // 4 channels * 9 taps per unrolled group
#define NGROUP    16                  // 16 groups * 36 = 576
#define GBYTES    147456u             // 36864 elements * 4B per group

// One workgroup (4 waves / 128 threads) per output spatial location.
// Per-location GEMM: D(16x16) = A(16x4 weight) x B(4x16 im2col), K = 576,
// 144 V_WMMA_F32_16X16X4_F32 per wave, fully unrolled; A loads use
// saddr + per-lane byte-offset VGPR + compile-time IOFFSET.
__global__ __launch_bounds__(128) void lc2d_wmma_f32(
    const float* __restrict__ x,    // (8,64,32,32)
    const float* __restrict__ w,    // (64,64,32,32,9), leading dim 1 dropped
    float* __restrict__ out)        // (8,64,32,32)
{
    // B matrix in K-pair layout: element (k, n) at (k>>1)*32 + n*2 + (k&1),
    // so the (K, K+1) pair a lane consumes is one aligned b64.
    __shared__ float ldsB[KDIM * 16];

    const int loc  = blockIdx.x;          // 0..1023 = oh*32+ow
    const int oh   = loc >> 5;
    const int ow   = loc & 31;
    const int tid  = threadIdx.x;
    const int lane = tid & 31;
    const int wave = tid >> 5;            // 0..3 -> cout tile of 16

    // ---- stage im2col B-matrix (pair layout) into LDS, b64 stores ----
    for (int e = tid; e < (KDIM / 2) * 16; e += 128) {
        const int p  = e >> 4;            // K pair 0..287
        const int n  = e & 15;            // batch column (pad 8..15 with 0)
        v2f pr = {0.0f, 0.0f};
        if (n < B_) {
#pragma unroll
            for (int h = 0; h < 2; ++h) {
                const int k  = (p << 1) + h;     // c-major K
                const int c  = k / KTAPS;
                const int kt = k - c * KTAPS;
                const int ih = oh - 1 + kt / 3;
                const int iw = ow - 1 + (kt - (kt / 3) * 3);
                if ((unsigned)ih < 32u && (unsigned)iw < 32u)
                    pr[h] = x[((n * CIN_ + c) << 10) + (ih << 5) + iw];
            }
        }
        *(v2f*)(ldsB + p * 32 + n * 2) = pr;
    }
    __syncthreads();

    // ---- per-lane constants ----
    const int m     = lane & 15;                  // M row (A) == N col (B)
    const int khalf = (lane >> 4) << 1;           // 0 | 2 per ISA f32 layouts
    const int o     = wave * 16 + m;

    // Per-lane A BYTE-offset base (weight = 151 MB, fits u32 / i32 > 0).
    const uint32_t awB =
        ((uint32_t)o * WSTRIDE_O + (uint32_t)loc * KTAPS) * 4u;

    // The (K/9, K%9) offset pattern repeats every 36 K: 18 per-lane byte
    // offsets precomputed once; the group stride is a compile-time IOFFSET.
    uint32_t bx[9], by[9];
#pragma unroll
    for (int s = 0; s < 9; ++s) {
        const int ka = 4 * s + khalf;             // K of .x within group
        const int kb = ka + 1;                    // K of .y
        const int ca = ka / KTAPS, ta = ka - KTAPS * ca;
        const int cb = kb / KTAPS, tb = kb - KTAPS * cb;
        bx[s] = awB + (uint32_t)(ca * WSTRIDE_C + ta) * 4u;
        by[s] = awB + (uint32_t)(cb * WSTRIDE_C + tb) * 4u;
    }

    const char* wB = (const char*)w;
    // Per-lane LDS base for B pairs: khalf and m folded in.
    const float* blane = ldsB + khalf * 16 + m * 2;

    // ---- fully unrolled WMMA accumulation: 16 groups x 9 steps ----
    v8f acc = {};
#pragma unroll
    for (int g = 0; g < NGROUP; ++g) {
#pragma unroll
        for (int s = 0; s < 9; ++s) {
            const int kg = g * GROUPK + 4 * s;    // even; khalf in bases

            v2f a, b;
            a.x = *(const float*)(wB + bx[s] + g * GBYTES);
            a.y = *(const float*)(wB + by[s] + g * GBYTES);
            b   = *(const v2f*)(blane + kg * 16); // ds_load b64, imm offset

            // (neg_a, A, neg_b, B, c_mod, C, reuse_a, reuse_b)
            acc = __builtin_amdgcn_wmma_f32_16x16x4_f32(
                      false, a, false, b, (short)0, acc, false, false);
        }
    }

    // ---- store D: lanes 0-15 -> rows v, lanes 16-31 -> rows v+8 ----
    if (m < B_) {
        const int mhi = (lane >> 4) << 3;
#pragma unroll
        for (int v = 0; v < 8; ++v) {
            const int oo = wave * 16 + v + mhi;
            out[((m * COUT_ + oo) << 10) + loc] = acc[v];
        }
    }
}

extern "C" void kernel_launch(void* const* d_in, const int* in_sizes, int n_in,
                              void* d_out, int out_size, void* d_ws, size_t ws_size,
                              hipStream_t stream) {
    (void)in_sizes; (void)n_in; (void)d_ws; (void)ws_size; (void)out_size;
    const float* x = (const float*)d_in[0];
    const float* w = (const float*)d_in[1];
    float* out = (float*)d_out;
    lc2d_wmma_f32<<<dim3(32 * 32), dim3(128), 0, stream>>>(x, w, out);
}